// GraphAttentionLayer_21749714387567
// MI455X (gfx1250) — compile-verified
//
#include <hip/hip_runtime.h>
#include <hip/hip_bf16.h>

typedef __attribute__((ext_vector_type(2))) float v2f;
typedef __attribute__((ext_vector_type(4))) float v4f;
typedef __attribute__((ext_vector_type(8))) float v8f;

#define IN_FEAT 256
#define HEADS 8
#define OUT_FEAT 32
#define HF 256           // HEADS*OUT_FEAT
#define LDS_STRIDE 260   // row stride (floats): 32 distinct banks per A-fragment read

// ---------------------------------------------------------------------------
// Kernel 1: h = x @ W^T  via V_WMMA_F32_16X16X4_F32 (fp32 matrix path).
// Block = 256 threads = 8 waves; each block computes one 16x256 tile of h.
// The 16x256 x-tile is staged through LDS once (b128 loads -> ds_store_b128),
// then each wave reads its A fragments with ds_load_b64 (no redundant global
// traffic, no divergent predicated loads in the K-loop).
// A (16x4 f32) per-lane: lanes 0-15 hold K={0,1}, lanes 16-31 hold K={2,3}.
// C/D (16x16 f32): VGPR r: lanes 0-15 -> M=r, N=lane; lanes 16-31 -> M=r+8.
// ---------------------------------------------------------------------------
__global__ __launch_bounds__(256) void gemm_h_wmma(
    const float* __restrict__ x, const float* __restrict__ W,
    float* __restrict__ h, int N) {
  __shared__ float lds_x[16 * LDS_STRIDE];  // 16.6 KB

  const int m0   = blockIdx.x * 16;
  const int tid  = threadIdx.x;
  const int lane = tid & 31;
  const int wave = tid >> 5;
  const int half = lane >> 4;   // 0: K pair {0,1}; 1: K pair {2,3}
  const int row  = lane & 15;   // M index (for A) / N index (for B, C)
  const int n0   = wave * 32;

  // ---- stage x tile into LDS: thread t loads 16 floats of row t>>4 ----
  {
    const int lrow = tid >> 4;                 // 0..15
    const int c0   = (tid & 15) * 16;          // 16-float chunk
    const int grow = min(m0 + lrow, N - 1);    // clamp: no divergent loads
    const float* __restrict__ src = x + (size_t)grow * IN_FEAT + c0;
    float* __restrict__ dst = &lds_x[lrow * LDS_STRIDE + c0];
#pragma unroll
    for (int i = 0; i < 4; ++i) {
      *(v4f*)(dst + 4 * i) = *(const v4f*)(src + 4 * i);
    }
  }
  __syncthreads();

  const float* __restrict__ w0 = W + (size_t)(n0 + row) * IN_FEAT;       // B[k,j]=W[j,k]
  const float* __restrict__ w1 = W + (size_t)(n0 + 16 + row) * IN_FEAT;
  const int aoff = row * LDS_STRIDE;

  v8f c0acc = {};
  v8f c1acc = {};

  for (int k4 = 0; k4 < IN_FEAT; k4 += 4) {
    const int kb = k4 + half * 2;
    v2f a;
    a.x = lds_x[aoff + kb + 0];   // merges to ds_load_b64 (8B aligned)
    a.y = lds_x[aoff + kb + 1];
    v2f b0, b1;
    b0.x = w0[kb + 0]; b0.y = w0[kb + 1];
    b1.x = w1[kb + 0]; b1.y = w1[kb + 1];
    c0acc = __builtin_amdgcn_wmma_f32_16x16x4_f32(false, a, false, b0,
                                                  (short)0, c0acc, false, false);
    c1acc = __builtin_amdgcn_wmma_f32_16x16x4_f32(false, a, false, b1,
                                                  (short)0, c1acc, false, false);
  }

#pragma unroll
  for (int r = 0; r < 8; ++r) {
    const int mm = m0 + r + 8 * half;
    if (mm < N) {
      h[(size_t)mm * HF + n0 + row]      = c0acc[r];
      h[(size_t)mm * HF + n0 + 16 + row] = c1acc[r];
    }
  }
}

// ---------------------------------------------------------------------------
// Kernel 2: per-node attention halves: s_src[n,h] = <h[n,h,:], a_src[h,:]>,
//           s_dst[n,h] = <h[n,h,:], a_dst[h,:]>. One thread per (node, head).
// ---------------------------------------------------------------------------
__global__ __launch_bounds__(256) void node_scores(
    const float* __restrict__ h, const float* __restrict__ a_src,
    const float* __restrict__ a_dst, float* __restrict__ s_src,
    float* __restrict__ s_dst, int N) {
  const int idx = blockIdx.x * blockDim.x + threadIdx.x;
  if (idx >= N * HEADS) return;
  const int head = idx & (HEADS - 1);
  const int n    = idx >> 3;
  const float* __restrict__ hp = h + (size_t)n * HF + head * OUT_FEAT;
  const float* __restrict__ as = a_src + head * OUT_FEAT;
  const float* __restrict__ ad = a_dst + head * OUT_FEAT;
  float ss = 0.0f, sd = 0.0f;
#pragma unroll
  for (int f = 0; f < OUT_FEAT; ++f) {
    const float v = hp[f];
    ss = fmaf(v, as[f], ss);
    sd = fmaf(v, ad[f], sd);
  }
  s_src[idx] = ss;
  s_dst[idx] = sd;
}

// ---------------------------------------------------------------------------
// Init kernels: out starts at bias (atomics accumulate into final answer),
// seg_max starts at -inf, seg_sum at 0.
// ---------------------------------------------------------------------------
__global__ __launch_bounds__(256) void init_out(
    float* __restrict__ out, const float* __restrict__ bias, int total) {
  const int idx = blockIdx.x * blockDim.x + threadIdx.x;
  if (idx < total) out[idx] = bias[idx & (HF - 1)];
}

__global__ __launch_bounds__(256) void init_seg(
    float* __restrict__ seg_max, float* __restrict__ seg_sum, int total) {
  const int idx = blockIdx.x * blockDim.x + threadIdx.x;
  if (idx < total) {
    seg_max[idx] = -__builtin_inff();
    seg_sum[idx] = 0.0f;
  }
}

// Float atomic max via signed-int max (val >= 0) / unsigned-int min (val < 0).
__device__ __forceinline__ void atomicMaxFloat(float* addr, float val) {
  if (val >= 0.0f) {
    atomicMax((int*)addr, __float_as_int(val));
  } else {
    atomicMin((unsigned int*)addr, __float_as_uint(val));
  }
}

// ---------------------------------------------------------------------------
// Kernel 3: per-(edge, head) raw attention + segment max over dst.
// ---------------------------------------------------------------------------
__global__ __launch_bounds__(256) void edge_alpha(
    const long long* __restrict__ ei, const float* __restrict__ s_src,
    const float* __restrict__ s_dst, float* __restrict__ alpha,
    float* __restrict__ seg_max, int E) {
  const int idx = blockIdx.x * blockDim.x + threadIdx.x;
  if (idx >= E * HEADS) return;
  const int head = idx & (HEADS - 1);
  const int e    = idx >> 3;
  const long long s = ei[e];
  const long long d = ei[(size_t)E + e];
  float a = s_src[(size_t)s * HEADS + head] + s_dst[(size_t)d * HEADS + head];
  a = (a >= 0.0f) ? a : 0.2f * a;  // LeakyReLU(0.2)
  alpha[idx] = a;
  atomicMaxFloat(&seg_max[(size_t)d * HEADS + head], a);
}

// ---------------------------------------------------------------------------
// Kernel 4: exp(alpha - seg_max[dst]) + segment sum over dst.
// ---------------------------------------------------------------------------
__global__ __launch_bounds__(256) void edge_exp(
    const long long* __restrict__ ei, float* __restrict__ alpha,
    const float* __restrict__ seg_max, float* __restrict__ seg_sum, int E) {
  const int idx = blockIdx.x * blockDim.x + threadIdx.x;
  if (idx >= E * HEADS) return;
  const int head = idx & (HEADS - 1);
  const int e    = idx >> 3;
  const long long d = ei[(size_t)E + e];
  const float m  = seg_max[(size_t)d * HEADS + head];
  const float ex = __expf(alpha[idx] - m);
  alpha[idx] = ex;
  atomicAdd(&seg_sum[(size_t)d * HEADS + head], ex);
}

// ---------------------------------------------------------------------------
// Kernel 4b: seg_sum -> 1/max(seg_sum, 1e-10)  (one divide per node-head,
// so the 6.4M-coefficient aggregation pass only multiplies).
// ---------------------------------------------------------------------------
__global__ __launch_bounds__(256) void seg_recip(
    float* __restrict__ seg_sum, int total) {
  const int idx = blockIdx.x * blockDim.x + threadIdx.x;
  if (idx < total) seg_sum[idx] = 1.0f / fmaxf(seg_sum[idx], 1e-10f);
}

// ---------------------------------------------------------------------------
// Kernel 5: weighted aggregation. One wave per edge; lane l handles feature
// head*32 + l for head = 0..7. All gathers are L2-resident (h = 51.2 MB).
// ---------------------------------------------------------------------------
__global__ __launch_bounds__(256) void edge_aggregate(
    const long long* __restrict__ ei, const float* __restrict__ h,
    const float* __restrict__ expa, const float* __restrict__ inv_sum,
    float* __restrict__ out, int E) {
  const int gwave = (int)((blockIdx.x * (size_t)blockDim.x + threadIdx.x) >> 5);
  const int lane  = threadIdx.x & 31;
  if (gwave >= E) return;
  const long long s = ei[gwave];
  const long long d = ei[(size_t)E + gwave];
  const float* __restrict__ hs = h + (size_t)s * HF;
  float* __restrict__ op = out + (size_t)d * HF;
#pragma unroll
  for (int head = 0; head < HEADS; ++head) {
    const float coeff = expa[(size_t)gwave * HEADS + head] *
                        inv_sum[(size_t)d * HEADS + head];
    const int j = head * OUT_FEAT + lane;
    atomicAdd(&op[j], hs[j] * coeff);
  }
}

// ---------------------------------------------------------------------------
// Launch
// ---------------------------------------------------------------------------
extern "C" void kernel_launch(void* const* d_in, const int* in_sizes, int n_in,
                              void* d_out, int out_size, void* d_ws, size_t ws_size,
                              hipStream_t stream) {
  const float*      x     = (const float*)d_in[0];
  const long long*  ei    = (const long long*)d_in[1];   // [2, E] int64
  const float*      W     = (const float*)d_in[2];       // [HF, IN_FEAT]
  const float*      a_src = (const float*)d_in[3];       // [H, F]
  const float*      a_dst = (const float*)d_in[4];       // [H, F]
  const float*      bias  = (const float*)d_in[5];       // [HF]

  const int N = in_sizes[0] / IN_FEAT;
  const int E = in_sizes[1] / 2;

  // Workspace layout (floats)
  float* h       = (float*)d_ws;                       // N * 256
  float* s_src   = h       + (size_t)N * HF;           // N * 8
  float* s_dst   = s_src   + (size_t)N * HEADS;        // N * 8
  float* seg_max = s_dst   + (size_t)N * HEADS;        // N * 8
  float* seg_sum = seg_max + (size_t)N * HEADS;        // N * 8
  float* alpha   = seg_sum + (size_t)N * HEADS;        // E * 8

  float* out = (float*)d_out;

  // 1. h = x @ W^T  (WMMA fp32, LDS-staged A tile)
  gemm_h_wmma<<<(N + 15) / 16, 256, 0, stream>>>(x, W, h, N);

  // 2. per-node score halves
  {
    const int total = N * HEADS;
    node_scores<<<(total + 255) / 256, 256, 0, stream>>>(h, a_src, a_dst,
                                                         s_src, s_dst, N);
  }

  // 3. inits
  {
    const int total = N * HF;
    init_out<<<(total + 255) / 256, 256, 0, stream>>>(out, bias, total);
  }
  {
    const int total = N * HEADS;
    init_seg<<<(total + 255) / 256, 256, 0, stream>>>(seg_max, seg_sum, total);
  }

  // 4. alpha + segment max
  {
    const int total = E * HEADS;
    edge_alpha<<<(total + 255) / 256, 256, 0, stream>>>(ei, s_src, s_dst,
                                                        alpha, seg_max, E);
  }

  // 5. exp + segment sum
  {
    const int total = E * HEADS;
    edge_exp<<<(total + 255) / 256, 256, 0, stream>>>(ei, alpha, seg_max,
                                                      seg_sum, E);
  }

  // 5b. reciprocal of segment sums
  {
    const int total = N * HEADS;
    seg_recip<<<(total + 255) / 256, 256, 0, stream>>>(seg_sum, total);
  }

  // 6. weighted aggregation (one wave per edge)
  {
    const size_t threads = (size_t)E * 32;
    edge_aggregate<<<(int)((threads + 255) / 256), 256, 0, stream>>>(
        ei, h, alpha, seg_sum, out, E);
  }
}